// get_model_52561809768872
// MI455X (gfx1250) — compile-verified
//
#include <hip/hip_runtime.h>

typedef __attribute__((ext_vector_type(2))) float v2f;
typedef __attribute__((ext_vector_type(8))) float v8f;

// ---------------------------------------------------------------------------
// WMMA f32 16x16x4 (fp32-exact matrix core path on gfx1250)
// ---------------------------------------------------------------------------
__device__ __forceinline__ v8f wmma_k4(v2f a, v2f b, v8f c) {
  return __builtin_amdgcn_wmma_f32_16x16x4_f32(false, a, false, b, (short)0, c,
                                               false, false);
}

// ---------------------------------------------------------------------------
// Transpose (B,4,N) -> pts (B,N,3) + norm (B,N)
// ---------------------------------------------------------------------------
__global__ void transpose_kernel(const float* __restrict__ xyz,
                                 float* __restrict__ pts,
                                 float* __restrict__ nrm, int N, int total) {
  int t = blockIdx.x * blockDim.x + threadIdx.x;
  if (t >= total) return;
  int b = t / N, n = t - b * N;
  const float* base = xyz + (size_t)b * 4 * N;
  pts[(size_t)t * 3 + 0] = base[0 * N + n];
  pts[(size_t)t * 3 + 1] = base[1 * N + n];
  pts[(size_t)t * 3 + 2] = base[2 * N + n];
  nrm[t] = base[3 * N + n];
}

// ---------------------------------------------------------------------------
// Fold conv-bias + batchnorm into per-channel (scale, shift):
//   y = relu(x*scale + shift),  scale = g*rsqrt(v+eps), shift = (bias-m)*scale + beta
// ---------------------------------------------------------------------------
__global__ void foldbn_kernel(const float* __restrict__ bias,
                              const float* __restrict__ g,
                              const float* __restrict__ beta,
                              const float* __restrict__ m,
                              const float* __restrict__ v,
                              float* __restrict__ scale,
                              float* __restrict__ shift, int C) {
  int o = blockIdx.x * blockDim.x + threadIdx.x;
  if (o >= C) return;
  float s = g[o] * rsqrtf(v[o] + 1e-5f);
  scale[o] = s;
  shift[o] = (bias[o] - m[o]) * s + beta[o];
}

// Pad weight rows from C to Cp columns (zero fill)
__global__ void padw_kernel(const float* __restrict__ w, float* __restrict__ wp,
                            int O, int C, int Cp) {
  int t = blockIdx.x * blockDim.x + threadIdx.x;
  if (t >= O * Cp) return;
  int o = t / Cp, c = t - o * Cp;
  wp[t] = (c < C) ? w[o * C + c] : 0.0f;
}

// ---------------------------------------------------------------------------
// Farthest point sampling. One block per batch. Exact argmax-first semantics.
// Dynamic LDS: px[N], py[N], pz[N], dist[N].
// ---------------------------------------------------------------------------
__global__ void fps_kernel(const float* __restrict__ xyz, int* __restrict__ out,
                           int N, int npoint) {
  extern __shared__ float sm[];
  float* px = sm;
  float* py = sm + N;
  float* pz = sm + 2 * N;
  float* dist = sm + 3 * N;
  __shared__ float rv[8];
  __shared__ int ri[8];
  __shared__ int sfar;
  int b = blockIdx.x;
  int tid = threadIdx.x;
  for (int i = tid; i < N; i += 256) {
    px[i] = xyz[(size_t)(b * N + i) * 3 + 0];
    py[i] = xyz[(size_t)(b * N + i) * 3 + 1];
    pz[i] = xyz[(size_t)(b * N + i) * 3 + 2];
    dist[i] = 1e10f;
  }
  __syncthreads();
  int far = 0;
  for (int j = 0; j < npoint; ++j) {
    if (tid == 0) out[b * npoint + j] = far;
    float cx = px[far], cy = py[far], cz = pz[far];
    float bv = -1.0f;
    int bi = 0x7fffffff;
    for (int i = tid; i < N; i += 256) {
      float dx = px[i] - cx, dy = py[i] - cy, dz = pz[i] - cz;
      float d = dx * dx + dy * dy + dz * dz;
      float nd = fminf(dist[i], d);
      dist[i] = nd;
      if (nd > bv) { bv = nd; bi = i; }  // strict > keeps lowest index per thread
    }
    for (int off = 16; off > 0; off >>= 1) {
      float ov = __shfl_down(bv, off, 32);
      int oi = __shfl_down(bi, off, 32);
      if (ov > bv || (ov == bv && oi < bi)) { bv = ov; bi = oi; }
    }
    if ((tid & 31) == 0) { rv[tid >> 5] = bv; ri[tid >> 5] = bi; }
    __syncthreads();
    if (tid == 0) {
      float bbv = rv[0];
      int bbi = ri[0];
      for (int w = 1; w < 8; ++w)
        if (rv[w] > bbv || (rv[w] == bbv && ri[w] < bbi)) { bbv = rv[w]; bbi = ri[w]; }
      sfar = bbi;
    }
    __syncthreads();
    far = sfar;
  }
}

// Gather sampled centroid coordinates: nxyz[t] = xyz[b, fidx[t]]
__global__ void gather_xyz_kernel(const float* __restrict__ xyz,
                                  const int* __restrict__ fidx,
                                  float* __restrict__ nxyz, int N, int S,
                                  int total) {
  int t = blockIdx.x * blockDim.x + threadIdx.x;
  if (t >= total) return;
  int b = t / S;
  int idx = fidx[t];
  nxyz[(size_t)t * 3 + 0] = xyz[(size_t)(b * N + idx) * 3 + 0];
  nxyz[(size_t)t * 3 + 1] = xyz[(size_t)(b * N + idx) * 3 + 1];
  nxyz[(size_t)t * 3 + 2] = xyz[(size_t)(b * N + idx) * 3 + 2];
}

// ---------------------------------------------------------------------------
// Ball query: one wave32 per centroid. Ordered compaction via ballot.
// Writes `rows` indices (first nsample qualifying, padded with first).
// ---------------------------------------------------------------------------
__global__ void ballq_kernel(const float* __restrict__ xyz,
                             const float* __restrict__ new_xyz,
                             int* __restrict__ gidx, int B, int N, int S,
                             int nsample, int rows, float r2) {
  int gw = (blockIdx.x * blockDim.x + threadIdx.x) >> 5;
  int lane = threadIdx.x & 31;
  if (gw >= B * S) return;
  int b = gw / S;
  float cx = new_xyz[(size_t)gw * 3 + 0];
  float cy = new_xyz[(size_t)gw * 3 + 1];
  float cz = new_xyz[(size_t)gw * 3 + 2];
  int cnt = 0;
  int firstIdx = 0;
  bool haveFirst = false;
  for (int n0 = 0; n0 < N && cnt < nsample; n0 += 32) {
    int n = n0 + lane;
    bool p = false;
    if (n < N) {
      float dx = xyz[(size_t)(b * N + n) * 3 + 0] - cx;
      float dy = xyz[(size_t)(b * N + n) * 3 + 1] - cy;
      float dz = xyz[(size_t)(b * N + n) * 3 + 2] - cz;
      p = (dx * dx + dy * dy + dz * dz) <= r2;
    }
    unsigned mask = (unsigned)__ballot(p);
    if (!haveFirst && mask != 0u) {
      firstIdx = n0 + __ffs(mask) - 1;
      haveFirst = true;
    }
    int before = __popc(mask & ((1u << lane) - 1u));
    int slot = cnt + before;
    if (p && slot < nsample) gidx[(size_t)gw * rows + slot] = n;
    cnt = min(nsample, cnt + __popc(mask));
  }
  for (int j = cnt + lane; j < rows; j += 32)
    gidx[(size_t)gw * rows + j] = firstIdx;
}

// ---------------------------------------------------------------------------
// One MLP layer via WMMA f32 16x16x4.
// X in LDS (rows x Cin), W global (Cout x Cin row-major, Cin % 4 == 0),
// Out in LDS (rows x Cout) = relu(X*W^T * scale + shift). blockDim = 128.
// ---------------------------------------------------------------------------
__device__ __forceinline__ void mlp_layer(
    const float* __restrict__ W, const float* __restrict__ scale,
    const float* __restrict__ shift, const float* Xl, float* Ol, int rows,
    int Cin, int Cout) {
  const int lane = threadIdx.x & 31;
  const int wave = threadIdx.x >> 5;
  const int mtiles = rows >> 4, ntiles = Cout >> 4, ktiles = Cin >> 2;
  const int total = mtiles * ntiles;
  const int half = lane >> 4;  // 0: K={0,1}, 1: K={2,3}
  const int l15 = lane & 15;
  for (int t = wave; t < total; t += 4) {
    int mt = t / ntiles, nt = t - mt * ntiles;
    v8f acc = {};
    const float* xp = Xl + (size_t)(mt * 16 + l15) * Cin + half * 2;
    const float* wp = W + (size_t)(nt * 16 + l15) * Cin + half * 2;
    for (int kt = 0; kt < ktiles; ++kt) {
      v2f a = *(const v2f*)(xp + kt * 4);
      v2f b = *(const v2f*)(wp + kt * 4);
      acc = wmma_k4(a, b, acc);
    }
    float sc = scale[nt * 16 + l15];
    float sh = shift[nt * 16 + l15];
    int rbase = mt * 16 + half * 8;
    int col = nt * 16 + l15;
#pragma unroll
    for (int v = 0; v < 8; ++v) {
      float y = fmaxf(acc[v] * sc + sh, 0.0f);
      Ol[(size_t)(rbase + v) * Cout + col] = y;
    }
  }
}

// ---------------------------------------------------------------------------
// Fused set-abstraction body: gather (rel-xyz ++ feats) -> MLP x3 -> max pool.
// One block (128 threads) per centroid. Dynamic LDS: ping/pong buffers.
// ---------------------------------------------------------------------------
__global__ void __launch_bounds__(128) sa_fused_kernel(
    const float* __restrict__ src_xyz, const float* __restrict__ src_feat,
    int F, const float* __restrict__ new_xyz, const int* __restrict__ gidx,
    int N, int S, int rows, int C0, int C1, int C2, int C3,
    const float* __restrict__ W1, const float* __restrict__ sc1,
    const float* __restrict__ sh1, const float* __restrict__ W2,
    const float* __restrict__ sc2, const float* __restrict__ sh2,
    const float* __restrict__ W3, const float* __restrict__ sc3,
    const float* __restrict__ sh3, float* __restrict__ out) {
  extern __shared__ float smem[];
  int bs = blockIdx.x;
  int b = bs / S;
  int tid = threadIdx.x;
  int c02 = (C0 > C2) ? C0 : C2;
  float* buf0 = smem;
  float* buf1 = smem + (size_t)rows * c02;

  float cx = new_xyz[(size_t)bs * 3 + 0];
  float cy = new_xyz[(size_t)bs * 3 + 1];
  float cz = new_xyz[(size_t)bs * 3 + 2];

  // Gather: rows x C0 = [rel_x, rel_y, rel_z, feats..., zero-pad]
  for (int e = tid; e < rows * C0; e += 128) {
    int r = e / C0, c = e - r * C0;
    int idx = gidx[(size_t)bs * rows + r];
    float val;
    if (c < 3) {
      float p = src_xyz[(size_t)(b * N + idx) * 3 + c];
      val = p - (c == 0 ? cx : (c == 1 ? cy : cz));
    } else if (c < 3 + F) {
      val = src_feat[(size_t)(b * N + idx) * F + (c - 3)];
    } else {
      val = 0.0f;
    }
    buf0[e] = val;
  }
  __syncthreads();
  mlp_layer(W1, sc1, sh1, buf0, buf1, rows, C0, C1);
  __syncthreads();
  mlp_layer(W2, sc2, sh2, buf1, buf0, rows, C1, C2);
  __syncthreads();
  mlp_layer(W3, sc3, sh3, buf0, buf1, rows, C2, C3);
  __syncthreads();
  // Max over rows (padded rows duplicate row 0 -> harmless)
  for (int c = tid; c < C3; c += 128) {
    float m = buf1[c];
    for (int r = 1; r < rows; ++r) m = fmaxf(m, buf1[(size_t)r * C3 + c]);
    out[(size_t)bs * C3 + c] = m;
  }
}

// ---------------------------------------------------------------------------
// Heads: ce = relu(bn(l3p @ wc^T)), ae = relu(bn(l3p @ wt^T))
// ---------------------------------------------------------------------------
__global__ void head_ce_ae_kernel(const float* __restrict__ l3p,
                                  const float* __restrict__ wc,
                                  const float* __restrict__ scc,
                                  const float* __restrict__ shc,
                                  const float* __restrict__ wt,
                                  const float* __restrict__ sct,
                                  const float* __restrict__ sht,
                                  float* __restrict__ ce,
                                  float* __restrict__ ae) {
  int t = blockIdx.x * blockDim.x + threadIdx.x;
  if (t >= 80 * 640) return;
  int bs = t / 640, o = t - bs * 640;
  const float* x = l3p + (size_t)bs * 512;
  const float* wrow;
  float sc, sh;
  if (o < 480) {
    wrow = wc + (size_t)o * 512; sc = scc[o]; sh = shc[o];
  } else {
    int o2 = o - 480;
    wrow = wt + (size_t)o2 * 512; sc = sct[o2]; sh = sht[o2];
  }
  float acc = 0.0f;
  for (int k = 0; k < 512; ++k) acc += x[k] * wrow[k];
  float y = fmaxf(acc * sc + sh, 0.0f);
  if (o < 480) ce[(size_t)bs * 480 + o] = y;
  else ae[(size_t)bs * 160 + (o - 480)] = y;
}

// Final outputs: d_out = [l3x (240) | coords (2400) | atoms (8000)]
__global__ void head_out_kernel(const float* __restrict__ l3x,
                                const float* __restrict__ ce,
                                const float* __restrict__ ae,
                                const float* __restrict__ w1,
                                const float* __restrict__ b1,
                                const float* __restrict__ w2,
                                const float* __restrict__ b2,
                                float* __restrict__ out) {
  int t = blockIdx.x * blockDim.x + threadIdx.x;
  if (t < 240) {
    out[t] = l3x[t];
    return;
  }
  if (t < 240 + 2400) {
    int u = t - 240;  // (bs, n, d)
    int bs = u / 30;
    int r = u - bs * 30;
    int n = r / 3, d = r - n * 3;
    const float* c = ce + (size_t)bs * 480 + n * 48 + d * 16;
    float acc = b1[0];
#pragma unroll
    for (int e = 0; e < 16; ++e) acc += c[e] * w1[e];
    out[t] = acc + l3x[bs * 3 + d];
    return;
  }
  if (t < 240 + 2400 + 800) {
    int u = t - 2640;  // (bs, n)
    int bs = u / 10, n = u - bs * 10;
    const float* a = ae + (size_t)bs * 160 + n * 16;
    float lg[10];
    float mx = -1e30f;
#pragma unroll
    for (int o = 0; o < 10; ++o) {
      float acc = b2[o];
#pragma unroll
      for (int e = 0; e < 16; ++e) acc += a[e] * w2[o * 16 + e];
      lg[o] = acc;
      mx = fmaxf(mx, acc);
    }
    float se = 0.0f;
#pragma unroll
    for (int o = 0; o < 10; ++o) se += __expf(lg[o] - mx);
    float lse = mx + __logf(se);
#pragma unroll
    for (int o = 0; o < 10; ++o) out[2640 + u * 10 + o] = lg[o] - lse;
  }
}

// ---------------------------------------------------------------------------
// Host launcher
// ---------------------------------------------------------------------------
extern "C" void kernel_launch(void* const* d_in, const int* in_sizes, int n_in,
                              void* d_out, int out_size, void* d_ws,
                              size_t ws_size, hipStream_t stream) {
  (void)in_sizes; (void)n_in; (void)out_size; (void)ws_size;
  const int B = 8, N1 = 4096;
  const int S1 = 1024, R1 = 64;  // nsample 60 -> pad 64
  const int S2 = 128, R2 = 32;   // nsample 20 -> pad 32
  const int S3 = 10, R3 = 16;    // nsample 8  -> pad 16

  const float* xyz_in = (const float*)d_in[0];

  // 9 SA conv layers + conv_coord + conv_type; each: (w, bias, g, beta, m, v)
  const float *W[11], *Bi[11], *Gg[11], *Be[11], *Mm[11], *Vv[11];
  const int Cc[11] = {64, 64, 128, 128, 128, 256, 256, 256, 512, 480, 160};
  int base = 1;
  for (int l = 0; l < 11; ++l) {
    W[l] = (const float*)d_in[base + 0];
    Bi[l] = (const float*)d_in[base + 1];
    Gg[l] = (const float*)d_in[base + 2];
    Be[l] = (const float*)d_in[base + 3];
    Mm[l] = (const float*)d_in[base + 4];
    Vv[l] = (const float*)d_in[base + 5];
    base += 6;
  }
  const float* w1 = (const float*)d_in[base + 0];  // fc1 w (1,16)
  const float* b1 = (const float*)d_in[base + 1];  // fc1 b (1)
  const float* w2 = (const float*)d_in[base + 2];  // fc2 w (10,16)
  const float* b2 = (const float*)d_in[base + 3];  // fc2 b (10)

  // Workspace carve
  float* f = (float*)d_ws;
  float* pts = f;   f += (size_t)B * N1 * 3;
  float* nrm = f;   f += (size_t)B * N1;
  float* l1x = f;   f += (size_t)B * S1 * 3;
  float* l1p = f;   f += (size_t)B * S1 * 128;
  float* l2x = f;   f += (size_t)B * S2 * 3;
  float* l2p = f;   f += (size_t)B * S2 * 256;
  float* l3x = f;   f += (size_t)B * S3 * 3;
  float* l3p = f;   f += (size_t)B * S3 * 512;
  float* ce = f;    f += (size_t)B * S3 * 480;
  float* ae = f;    f += (size_t)B * S3 * 160;
  float* wpad2 = f; f += 128 * 132;
  float* wpad3 = f; f += 256 * 260;
  float *SC[11], *SH[11];
  for (int l = 0; l < 11; ++l) { SC[l] = f; f += Cc[l]; SH[l] = f; f += Cc[l]; }
  int* ip = (int*)f;
  int* fidx1 = ip; ip += B * S1;
  int* gidx1 = ip; ip += (size_t)B * S1 * R1;
  int* fidx2 = ip; ip += B * S2;
  int* gidx2 = ip; ip += B * S2 * R2;
  int* fidx3 = ip; ip += B * S3;
  int* gidx3 = ip; ip += B * S3 * R3;

  // Prep
  transpose_kernel<<<(B * N1 + 255) / 256, 256, 0, stream>>>(xyz_in, pts, nrm,
                                                             N1, B * N1);
  for (int l = 0; l < 11; ++l)
    foldbn_kernel<<<(Cc[l] + 255) / 256, 256, 0, stream>>>(
        Bi[l], Gg[l], Be[l], Mm[l], Vv[l], SC[l], SH[l], Cc[l]);
  padw_kernel<<<(128 * 132 + 255) / 256, 256, 0, stream>>>(W[3], wpad2, 128,
                                                           131, 132);
  padw_kernel<<<(256 * 260 + 255) / 256, 256, 0, stream>>>(W[6], wpad3, 256,
                                                           259, 260);

  // --- SA1: 4096 -> 1024, r=0.1, nsample=60, C: 4 -> 64 -> 64 -> 128
  fps_kernel<<<B, 256, 16 * N1, stream>>>(pts, fidx1, N1, S1);
  gather_xyz_kernel<<<(B * S1 + 255) / 256, 256, 0, stream>>>(pts, fidx1, l1x,
                                                              N1, S1, B * S1);
  ballq_kernel<<<(B * S1 + 3) / 4, 128, 0, stream>>>(pts, l1x, gidx1, B, N1,
                                                     S1, 60, R1, 0.01f);
  sa_fused_kernel<<<B * S1, 128, (size_t)R1 * (64 + 128) * 4, stream>>>(
      pts, nrm, 1, l1x, gidx1, N1, S1, R1, 4, 64, 64, 128, W[0], SC[0], SH[0],
      W[1], SC[1], SH[1], W[2], SC[2], SH[2], l1p);

  // --- SA2: 1024 -> 128, r=0.3, nsample=20, C: 132(pad) -> 128 -> 128 -> 256
  fps_kernel<<<B, 256, 16 * S1, stream>>>(l1x, fidx2, S1, S2);
  gather_xyz_kernel<<<(B * S2 + 255) / 256, 256, 0, stream>>>(l1x, fidx2, l2x,
                                                              S1, S2, B * S2);
  ballq_kernel<<<(B * S2 + 3) / 4, 128, 0, stream>>>(l1x, l2x, gidx2, B, S1,
                                                     S2, 20, R2, 0.09f);
  sa_fused_kernel<<<B * S2, 128, (size_t)R2 * (132 + 256) * 4, stream>>>(
      l1x, l1p, 128, l2x, gidx2, S1, S2, R2, 132, 128, 128, 256, wpad2, SC[3],
      SH[3], W[4], SC[4], SH[4], W[5], SC[5], SH[5], l2p);

  // --- SA3: 128 -> 10, r=0.5, nsample=8, C: 260(pad) -> 256 -> 256 -> 512
  fps_kernel<<<B, 256, 16 * S2, stream>>>(l2x, fidx3, S2, S3);
  gather_xyz_kernel<<<(B * S3 + 255) / 256, 256, 0, stream>>>(l2x, fidx3, l3x,
                                                              S2, S3, B * S3);
  ballq_kernel<<<(B * S3 + 3) / 4, 128, 0, stream>>>(l2x, l3x, gidx3, B, S2,
                                                     S3, 8, R3, 0.25f);
  sa_fused_kernel<<<B * S3, 128, (size_t)R3 * (260 + 512) * 4, stream>>>(
      l2x, l2p, 256, l3x, gidx3, S2, S3, R3, 260, 256, 256, 512, wpad3, SC[6],
      SH[6], W[7], SC[7], SH[7], W[8], SC[8], SH[8], l3p);

  // --- Heads
  head_ce_ae_kernel<<<(80 * 640 + 255) / 256, 256, 0, stream>>>(
      l3p, W[9], SC[9], SH[9], W[10], SC[10], SH[10], ce, ae);
  head_out_kernel<<<(3440 + 255) / 256, 256, 0, stream>>>(
      l3x, ce, ae, w1, b1, w2, b2, (float*)d_out);
}